// ColorHistograms_37134287241740
// MI455X (gfx1250) — compile-verified
//
#include <hip/hip_runtime.h>
#include <hip/hip_bf16.h>

typedef __attribute__((ext_vector_type(2))) float v2f;
typedef __attribute__((ext_vector_type(8))) float v8f;

#define T_DIM 2048
#define B_DIM 8
#define WIN   101
#define HALF  50

// ---------------------------------------------------------------------------
// Phase 1: mean over H*W (4096 pixels x 3 channels) per (b,t).
// One block per (b,t) frame; 256 threads stream 3072 float4s (48 KB) with
// coalesced b128 loads (12-deep unroll = 12 outstanding loads/thread for MLP).
// Channel of v.x at float4-index i4 is (4*i4) % 3; v.w shares it.
// HBM-bound: 768 MiB total -> ~35 us floor at 23.3 TB/s.
// ---------------------------------------------------------------------------
__global__ __launch_bounds__(256) void ch_mean_kernel(const float* __restrict__ frames,
                                                      float4* __restrict__ means) {
  const int bt  = blockIdx.x;                       // b*T_DIM + t
  const int tid = threadIdx.x;
  const float4* __restrict__ fp =
      reinterpret_cast<const float4*>(frames) + (size_t)bt * 3072;

  float r = 0.f, g = 0.f, b = 0.f;
#pragma unroll
  for (int it = 0; it < 12; ++it) {
    const int i4 = tid + it * 256;
    const float4 v = fp[i4];
    const int k = (i4 * 4) % 3;      // channel of v.x and v.w
    const float s0 = v.x + v.w;      // both land on channel k
    // v.y -> channel (k+1)%3, v.z -> channel (k+2)%3
    r += (k == 0) ? s0 : (k == 1) ? v.z : v.y;
    g += (k == 1) ? s0 : (k == 2) ? v.z : v.y;
    b += (k == 2) ? s0 : (k == 0) ? v.z : v.y;
  }

  __shared__ float4 sm[256];
  sm[tid] = make_float4(r, g, b, 0.f);
  __syncthreads();
#pragma unroll
  for (int s = 128; s > 0; s >>= 1) {
    if (tid < s) {
      float4 o = sm[tid + s];
      float4 m = sm[tid];
      m.x += o.x; m.y += o.y; m.z += o.z;
      sm[tid] = m;
    }
    __syncthreads();
  }
  if (tid == 0) {
    const float inv = 1.0f / 4096.0f;
    float4 m = sm[0];
    means[bt] = make_float4(m.x * inv, m.y * inv, m.z * inv, 0.f);
  }
}

// ---------------------------------------------------------------------------
// Phase 2: windowed color similarity via WMMA Gram matrices.
//   diff(t,a) = |c_t|^2 + |c_a|^2 - 2 * (c_t . c_a)
// The dot-product tile (16 centers x 16 absolute neighbors, K=4 = RGB+pad) is
// one V_WMMA_F32_16X16X4_F32. Output coordinate j = a - max(0, t-50); each
// (t,j) maps to exactly one absolute index a in [A0, A0+128), so the 8 waves
// (one 16-wide a-tile each) cover every output element exactly once.
// ---------------------------------------------------------------------------
__global__ __launch_bounds__(256) void sim_wmma_kernel(const float4* __restrict__ means,
                                                       float* __restrict__ out) {
  const int blk = blockIdx.x;
  const int b   = blk >> 7;        // / (T_DIM/16)
  const int tt  = blk & 127;
  const int t0  = tt * 16;
  const float4* __restrict__ mc = means + (size_t)b * T_DIM;

  const int tid  = threadIdx.x;
  const int wave = tid >> 5;       // 0..7 -> which a-tile
  const int lane = tid & 31;
  const int m    = lane & 15;
  const int hi   = lane >> 4;      // 0: K={0,1}; 1: K={2,3}

  __shared__ float normC[16];
  if (tid < 16) {
    const float4 c = mc[t0 + tid];
    normC[tid] = c.x * c.x + c.y * c.y + c.z * c.z;
  }
  __syncthreads();

  // A-matrix (16x4 f32): lane m holds row M=m; VGPR0 = K=(hi?2:0), VGPR1 = K=(hi?3:1)
  const float4 c = mc[t0 + m];
  v2f A;
  A.x = hi ? c.z : c.x;
  A.y = hi ? 0.f : c.y;

  // B-matrix (4x16 f32): lane m holds column N=m of absolute-index tile
  const int A0 = max(0, t0 - HALF);
  const int a  = A0 + wave * 16 + m;
  const int ac = min(a, T_DIM - 1);          // clamped gather (masked if OOB)
  const float4 n = mc[ac];
  v2f Bv;
  Bv.x = hi ? n.z : n.x;
  Bv.y = hi ? 0.f : n.y;
  const float nn = n.x * n.x + n.y * n.y + n.z * n.z;

  // D[M][N] = sum_k A[M][k] * B[k][N]  (rank-4 f32 Gram tile)
  v8f D = {};
  D = __builtin_amdgcn_wmma_f32_16x16x4_f32(
      /*neg_a=*/false, A, /*neg_b=*/false, Bv,
      /*c_mod=*/(short)0, D, /*reuse_a=*/false, /*reuse_b=*/false);

  // Scatter: lane holds D rows (v + 8*hi), column m.
#pragma unroll
  for (int v = 0; v < 8; ++v) {
    const int row   = v + 8 * hi;
    const int t     = t0 + row;
    const int start = max(0, t - HALF);
    const int j     = a - start;
    if ((unsigned)j < (unsigned)WIN) {
      const bool valid = (a < T_DIM) && (a < t + HALF + 1);
      const float dot  = D[v];
      float diff = normC[row] + nn - 2.0f * dot;
      diff = fmaxf(diff, 0.0f);              // guard FP cancellation
      // v_rcp_f32 (~1 ulp) instead of IEEE div sequence: accuracy limit here
      // is the f32 cancellation in `diff`, not the reciprocal.
      const float sim = valid ? __builtin_amdgcn_rcpf(1.0f + diff) : 0.0f;
      out[((size_t)b * T_DIM + t) * WIN + j] = sim;
    }
  }
}

extern "C" void kernel_launch(void* const* d_in, const int* in_sizes, int n_in,
                              void* d_out, int out_size, void* d_ws, size_t ws_size,
                              hipStream_t stream) {
  (void)in_sizes; (void)n_in; (void)out_size; (void)ws_size;
  const float* frames = (const float*)d_in[0];   // (8, 2048, 64, 64, 3) f32
  float4* means = (float4*)d_ws;                 // (8*2048) float4 = 256 KB scratch
  float* out    = (float*)d_out;                 // (8, 2048, 101) f32

  ch_mean_kernel<<<B_DIM * T_DIM, 256, 0, stream>>>(frames, means);
  sim_wmma_kernel<<<B_DIM * (T_DIM / 16), 256, 0, stream>>>(means, out);
}